// PhoenixGNN_47966194762008
// MI455X (gfx1250) — compile-verified
//
#include <hip/hip_runtime.h>
#include <math.h>

typedef __attribute__((ext_vector_type(2))) float v2f;
typedef __attribute__((ext_vector_type(8))) float v8f;

#define NN 50000
#define NE 1000000
#define IN_CH 256
#define HID 128
#define HEADS 2
#define OUTC 3
#define F1 (HEADS * HID)   // 256
#define NEG_SLOPE 0.2f

// ---------- helpers ----------
__device__ inline float wave_sum(float v) {
#pragma unroll
  for (int off = 16; off > 0; off >>= 1) v += __shfl_xor(v, off, 32);
  return v;
}

// monotone float <-> uint encoding for atomic max on floats
__device__ inline unsigned f2ord(float f) {
  unsigned b = __float_as_uint(f);
  return (b & 0x80000000u) ? ~b : (b | 0x80000000u);
}
__device__ inline float ord2f(unsigned u) {
  return (u & 0x80000000u) ? __uint_as_float(u & 0x7FFFFFFFu) : __uint_as_float(~u);
}

__device__ inline float leaky(float v) { return v > 0.f ? v : NEG_SLOPE * v; }

// ---------- GEMM1: h1 = x @ W1 via V_WMMA_F32_16X16X4_F32 ----------
// one wave per (16-row tile) x (64-col group): 4 accumulators reuse one A frag
__global__ void k_gemm1_wmma(const float* __restrict__ x, const float* __restrict__ W,
                             float* __restrict__ h1) {
  int wid  = (blockIdx.x * blockDim.x + threadIdx.x) >> 5;
  int lane = threadIdx.x & 31;
  const int NGROUPS = F1 / 64;            // 4 column groups of 4 tiles
  int mtile = wid / NGROUPS;
  int ng    = wid % NGROUPS;
  if (mtile >= NN / 16) return;

  int row     = mtile * 16 + (lane & 15);   // A: M = lane%16
  int colbase = ng * 64 + (lane & 15);      // B/D: N = colbase + 16*j
  int khalf   = (lane >> 4) << 1;           // lanes 0-15: K+0/1, lanes 16-31: K+2/3

  v8f acc0 = {}, acc1 = {}, acc2 = {}, acc3 = {};
  const float* xrow = x + (size_t)row * IN_CH;

  for (int k0 = 0; k0 < IN_CH; k0 += 4) {
    int kk = k0 + khalf;
    v2f a = *(const v2f*)(xrow + kk);                 // K, K+1 (8B aligned)
    const float* wr0 = W + (size_t)kk * F1 + colbase;
    const float* wr1 = wr0 + F1;
    v2f b0, b1, b2, b3;
    b0[0] = wr0[0];  b0[1] = wr1[0];
    b1[0] = wr0[16]; b1[1] = wr1[16];
    b2[0] = wr0[32]; b2[1] = wr1[32];
    b3[0] = wr0[48]; b3[1] = wr1[48];
    acc0 = __builtin_amdgcn_wmma_f32_16x16x4_f32(false, a, false, b0, (short)0, acc0, false, false);
    acc1 = __builtin_amdgcn_wmma_f32_16x16x4_f32(false, a, false, b1, (short)0, acc1, false, false);
    acc2 = __builtin_amdgcn_wmma_f32_16x16x4_f32(false, a, false, b2, (short)0, acc2, false, false);
    acc3 = __builtin_amdgcn_wmma_f32_16x16x4_f32(false, a, false, b3, (short)0, acc3, false, false);
  }
  // D layout: VGPR v holds M = v + (lane>=16 ? 8 : 0), N = lane%16
  int rbase = mtile * 16 + ((lane >> 4) << 3);
#pragma unroll
  for (int v = 0; v < 8; ++v) {
    float* drow = h1 + (size_t)(rbase + v) * F1 + colbase;
    drow[0]  = acc0[v];
    drow[16] = acc1[v];
    drow[32] = acc2[v];
    drow[48] = acc3[v];
  }
}

// ---------- attention logits layer 1: als/ald [NN*HEADS] ----------
__global__ void k_attn1(const float* __restrict__ h1, const float* __restrict__ a_s,
                        const float* __restrict__ a_d, float* __restrict__ als,
                        float* __restrict__ ald) {
  int wid  = (blockIdx.x * blockDim.x + threadIdx.x) >> 5;
  int lane = threadIdx.x & 31;
  if (wid >= NN * HEADS) return;
  int node = wid >> 1, head = wid & 1;
  const float* hp = h1 + (size_t)node * F1 + head * HID;
  const float* as = a_s + head * HID;
  const float* ad = a_d + head * HID;
  float ss = 0.f, sd = 0.f;
#pragma unroll
  for (int i = 0; i < HID / 32; ++i) {
    int c = lane + 32 * i;
    float hv = hp[c];
    ss += hv * as[c];
    sd += hv * ad[c];
  }
  ss = wave_sum(ss);
  sd = wave_sum(sd);
  if (lane == 0) { als[wid] = ss; ald[wid] = sd; }
}

// ---------- edge segment-max (ordered-uint atomicMax), H heads ----------
__global__ void k_edge_max(const long long* __restrict__ ei, const float* __restrict__ als,
                           const float* __restrict__ ald, unsigned* __restrict__ nmax,
                           int heads) {
  int t = blockIdx.x * blockDim.x + threadIdx.x;
  if (t >= NE * heads) return;
  int e = t / heads, h = t % heads;
  int s = (int)ei[e], d = (int)ei[NE + e];
  float v = leaky(als[s * heads + h] + ald[d * heads + h]);
  atomicMax(&nmax[d * heads + h], f2ord(v));
}

// ---------- edge weights + denom ----------
__global__ void k_edge_w(const long long* __restrict__ ei, const float* __restrict__ als,
                         const float* __restrict__ ald, const unsigned* __restrict__ nmax,
                         float* __restrict__ w, float* __restrict__ den, int heads) {
  int t = blockIdx.x * blockDim.x + threadIdx.x;
  if (t >= NE * heads) return;
  int e = t / heads, h = t % heads;
  int s = (int)ei[e], d = (int)ei[NE + e];
  float v  = leaky(als[s * heads + h] + ald[d * heads + h]);
  float m  = ord2f(nmax[d * heads + h]);
  float wv = __expf(v - m);
  w[t] = wv;
  atomicAdd(&den[d * heads + h], wv);
}

// ---------- layer-1 aggregation: wave per edge, b128 gather + f32 atomics ----------
__global__ void k_aggr1(const long long* __restrict__ ei, const float* __restrict__ h1,
                        const float* __restrict__ w, const float* __restrict__ den,
                        float* __restrict__ out1) {
  int wid  = (blockIdx.x * blockDim.x + threadIdx.x) >> 5;
  int lane = threadIdx.x & 31;
  if (wid >= NE) return;
  int s = (int)ei[wid], d = (int)ei[NE + wid];
  float c0 = w[wid * 2 + 0] / fmaxf(den[d * 2 + 0], 1e-16f);
  float c1 = w[wid * 2 + 1] / fmaxf(den[d * 2 + 1], 1e-16f);
  const float4* hp = (const float4*)(h1 + (size_t)s * F1);
  float* op = out1 + (size_t)d * F1;
#pragma unroll
  for (int i = 0; i < 2; ++i) {               // i=0 -> head 0 (c<128), i=1 -> head 1
    int q = lane + 32 * i;                    // float4 index, coalesced per wave
    float4 v = hp[q];
    float cc = (i == 0) ? c0 : c1;
    atomicAdd(&op[q * 4 + 0], v.x * cc);
    atomicAdd(&op[q * 4 + 1], v.y * cc);
    atomicAdd(&op[q * 4 + 2], v.z * cc);
    atomicAdd(&op[q * 4 + 3], v.w * cc);
  }
}

// ---------- bias + ELU in place on out1 (float4) ----------
__global__ void k_elu1(float* __restrict__ out1, const float* __restrict__ b1) {
  int t = blockIdx.x * blockDim.x + threadIdx.x;   // float4 index
  if (t >= NN * F1 / 4) return;
  float4 v = ((const float4*)out1)[t];
  float4 b = ((const float4*)b1)[t & (F1 / 4 - 1)];
  v.x += b.x; v.y += b.y; v.z += b.z; v.w += b.w;
  v.x = v.x > 0.f ? v.x : expm1f(v.x);
  v.y = v.y > 0.f ? v.y : expm1f(v.y);
  v.z = v.z > 0.f ? v.z : expm1f(v.z);
  v.w = v.w > 0.f ? v.w : expm1f(v.w);
  ((float4*)out1)[t] = v;
}

// ---------- GEMM2 (256 -> 3) + layer-2 logits, wave per node ----------
__global__ void k_gemm2(const float* __restrict__ h, const float* __restrict__ W2,
                        const float* __restrict__ as2, const float* __restrict__ ad2,
                        float* __restrict__ h2, float* __restrict__ als2,
                        float* __restrict__ ald2) {
  int wid  = (blockIdx.x * blockDim.x + threadIdx.x) >> 5;
  int lane = threadIdx.x & 31;
  if (wid >= NN) return;
  const float4* hp = (const float4*)(h + (size_t)wid * F1);
  float s0 = 0.f, s1 = 0.f, s2 = 0.f;
#pragma unroll
  for (int i = 0; i < 2; ++i) {
    int q = lane + 32 * i;                    // float4 index, coalesced
    float4 hv = hp[q];
    const float* wp = W2 + q * 4 * OUTC;
    s0 += hv.x * wp[0] + hv.y * wp[3] + hv.z * wp[6] + hv.w * wp[9];
    s1 += hv.x * wp[1] + hv.y * wp[4] + hv.z * wp[7] + hv.w * wp[10];
    s2 += hv.x * wp[2] + hv.y * wp[5] + hv.z * wp[8] + hv.w * wp[11];
  }
  s0 = wave_sum(s0); s1 = wave_sum(s1); s2 = wave_sum(s2);
  if (lane == 0) {
    h2[wid * OUTC + 0] = s0;
    h2[wid * OUTC + 1] = s1;
    h2[wid * OUTC + 2] = s2;
    als2[wid] = s0 * as2[0] + s1 * as2[1] + s2 * as2[2];
    ald2[wid] = s0 * ad2[0] + s1 * ad2[1] + s2 * ad2[2];
  }
}

// ---------- layer-2 aggregation: thread per edge (3 channels) ----------
__global__ void k_aggr2(const long long* __restrict__ ei, const float* __restrict__ h2,
                        const float* __restrict__ w, const float* __restrict__ den,
                        float* __restrict__ out2) {
  int e = blockIdx.x * blockDim.x + threadIdx.x;
  if (e >= NE) return;
  int s = (int)ei[e], d = (int)ei[NE + e];
  float c = w[e] / fmaxf(den[d], 1e-16f);
#pragma unroll
  for (int j = 0; j < OUTC; ++j)
    atomicAdd(&out2[d * OUTC + j], h2[s * OUTC + j] * c);
}

// ---------- bias + log_softmax, thread per node ----------
__global__ void k_final(const float* __restrict__ out2, const float* __restrict__ b2,
                        float* __restrict__ out) {
  int n = blockIdx.x * blockDim.x + threadIdx.x;
  if (n >= NN) return;
  float v0 = out2[n * OUTC + 0] + b2[0];
  float v1 = out2[n * OUTC + 1] + b2[1];
  float v2 = out2[n * OUTC + 2] + b2[2];
  float m  = fmaxf(v0, fmaxf(v1, v2));
  float ls = m + logf(__expf(v0 - m) + __expf(v1 - m) + __expf(v2 - m));
  out[n * OUTC + 0] = v0 - ls;
  out[n * OUTC + 1] = v1 - ls;
  out[n * OUTC + 2] = v2 - ls;
}

extern "C" void kernel_launch(void* const* d_in, const int* in_sizes, int n_in,
                              void* d_out, int out_size, void* d_ws, size_t ws_size,
                              hipStream_t stream) {
  const float*     x    = (const float*)d_in[0];
  const long long* ei   = (const long long*)d_in[1];
  const float*     W1   = (const float*)d_in[2];
  const float*     asr1 = (const float*)d_in[3];
  const float*     ads1 = (const float*)d_in[4];
  const float*     b1   = (const float*)d_in[5];
  const float*     W2   = (const float*)d_in[6];
  const float*     asr2 = (const float*)d_in[7];
  const float*     ads2 = (const float*)d_in[8];
  const float*     b2   = (const float*)d_in[9];
  float* out = (float*)d_out;

  // ---- workspace carve (floats / uints; h1/out1 rows stay 16B aligned) ----
  float* ws = (float*)d_ws;
  float*    h1    = ws;            ws += (size_t)NN * F1;    // 12.8M
  float*    out1  = ws;            ws += (size_t)NN * F1;    // 12.8M
  float*    als1  = ws;            ws += (size_t)NN * HEADS;
  float*    ald1  = ws;            ws += (size_t)NN * HEADS;
  unsigned* nmax1 = (unsigned*)ws; ws += (size_t)NN * HEADS;
  float*    den1  = ws;            ws += (size_t)NN * HEADS;
  float*    w1    = ws;            ws += (size_t)NE * HEADS; // 2M
  float*    h2    = ws;            ws += (size_t)NN * OUTC;
  float*    als2  = ws;            ws += (size_t)NN;
  float*    ald2  = ws;            ws += (size_t)NN;
  unsigned* nmax2 = (unsigned*)ws; ws += (size_t)NN;
  float*    den2  = ws;            ws += (size_t)NN;
  float*    w2    = ws;            ws += (size_t)NE;          // 1M
  float*    out2  = ws;            ws += (size_t)NN * OUTC;

  // ---- per-call zero init (graph-capture safe) ----
  hipMemsetAsync(out1,  0, (size_t)NN * F1 * 4, stream);
  hipMemsetAsync(nmax1, 0, (size_t)NN * HEADS * 4, stream);  // 0 == ordered(-inf)
  hipMemsetAsync(den1,  0, (size_t)NN * HEADS * 4, stream);
  hipMemsetAsync(nmax2, 0, (size_t)NN * 4, stream);
  hipMemsetAsync(den2,  0, (size_t)NN * 4, stream);
  hipMemsetAsync(out2,  0, (size_t)NN * OUTC * 4, stream);

  const int BT = 256;  // 8 wave32s per block

  // layer 1
  {
    int waves = (NN / 16) * (F1 / 64);  // 12500 waves, 4 tiles each
    k_gemm1_wmma<<<(waves * 32 + BT - 1) / BT, BT, 0, stream>>>(x, W1, h1);
  }
  {
    int waves = NN * HEADS;
    k_attn1<<<(waves * 32 + BT - 1) / BT, BT, 0, stream>>>(h1, asr1, ads1, als1, ald1);
  }
  {
    int th = NE * HEADS;
    k_edge_max<<<(th + BT - 1) / BT, BT, 0, stream>>>(ei, als1, ald1, nmax1, HEADS);
    k_edge_w<<<(th + BT - 1) / BT, BT, 0, stream>>>(ei, als1, ald1, nmax1, w1, den1, HEADS);
  }
  {
    long long th = (long long)NE * 32;
    k_aggr1<<<(int)((th + BT - 1) / BT), BT, 0, stream>>>(ei, h1, w1, den1, out1);
  }
  {
    int th = NN * F1 / 4;
    k_elu1<<<(th + BT - 1) / BT, BT, 0, stream>>>(out1, b1);
  }

  // layer 2
  {
    int waves = NN;
    k_gemm2<<<(waves * 32 + BT - 1) / BT, BT, 0, stream>>>(out1, W2, asr2, ads2, h2, als2, ald2);
  }
  {
    k_edge_max<<<(NE + BT - 1) / BT, BT, 0, stream>>>(ei, als2, ald2, nmax2, 1);
    k_edge_w<<<(NE + BT - 1) / BT, BT, 0, stream>>>(ei, als2, ald2, nmax2, w2, den2, 1);
    k_aggr2<<<(NE + BT - 1) / BT, BT, 0, stream>>>(ei, h2, w2, den2, out2);
  }
  k_final<<<(NN + BT - 1) / BT, BT, 0, stream>>>(out2, b2, out);
}